// SAGEAggregator_26465588478211
// MI455X (gfx1250) — compile-verified
//
#include <hip/hip_runtime.h>

// ---------------------------------------------------------------------------
// Fused GraphSAGE layer for MI455X (gfx1250, wave32):
//   out[n,o] = sum_c x[n,c]*W_l[o,c] + sum_c mean_k(neigh[n,k,c])*W_r[o,c]
//              + b_l[o] + b_r[o]
// One block = one 16-row slab of N. Phase 1 streams the slab's neigh data
// (the HBM-bound part, ~1 MB/block, non-temporal) and reduces the mean into
// LDS. Phase 2 runs fp32 WMMA (v_wmma_f32_16x16x4_f32) over K=C=512 with two
// independent accumulators so consecutive WMMAs don't serialize on one D.
// ---------------------------------------------------------------------------

typedef __attribute__((ext_vector_type(2))) float v2f;
typedef __attribute__((ext_vector_type(4))) float v4f;
typedef __attribute__((ext_vector_type(8))) float v8f;

#define C_DIM   512
#define K_NEIGH 32
#define OUT_DIM 512
#define ROWS    16
// LDS row stride in floats, padded so that the A-fragment ds_load_b64 pattern
// (16 lanes at stride LDS_STRIDE, +2 floats for lanes 16-31) covers all 64
// banks with no conflict: 516 mod 64(words) = 4 -> lane L hits banks {4L,4L+1}.
#define LDS_STRIDE 516

__global__ __launch_bounds__(256)
void sage_fused_kernel(const float* __restrict__ x,
                       const float* __restrict__ neigh,
                       const float* __restrict__ Wl,
                       const float* __restrict__ bl,
                       const float* __restrict__ Wr,
                       const float* __restrict__ br,
                       float* __restrict__ out)
{
    __shared__ float x_lds[ROWS * LDS_STRIDE];
    __shared__ float g_lds[ROWS * LDS_STRIDE];

    const int n0  = blockIdx.x * ROWS;
    const int tid = threadIdx.x;

    // ---------------- Phase 1a: load x tile (16 x 512 fp32) ----------------
    // 2048 vec4 elements, 256 threads -> 8 per thread, fully coalesced.
    for (int idx = tid; idx < ROWS * (C_DIM / 4); idx += 256) {
        const int m  = idx >> 7;     // row in tile (idx / 128)
        const int c4 = idx & 127;    // vec4 column
        v4f v = ((const v4f*)(x + (size_t)(n0 + m) * C_DIM))[c4];
        ((v4f*)(x_lds + m * LDS_STRIDE))[c4] = v;
        // Warm the cache path for the neighbor rows this thread streams next.
        __builtin_prefetch(neigh + (size_t)(n0 + m) * K_NEIGH * C_DIM + 4 * c4, 0, 0);
    }

    // ---------------- Phase 1b: agg = mean over K neighbors ----------------
    // 1 MB single-use stream per block: non-temporal loads keep it from
    // evicting the weight working set in WGP$/L2. Consecutive threads read
    // consecutive vec4s -> fully coalesced 128B lines.
    for (int idx = tid; idx < ROWS * (C_DIM / 4); idx += 256) {
        const int m  = idx >> 7;
        const int c4 = idx & 127;
        const v4f* base =
            (const v4f*)(neigh + (size_t)(n0 + m) * K_NEIGH * C_DIM) + c4;
        v4f s = {0.f, 0.f, 0.f, 0.f};
        #pragma unroll 8
        for (int k = 0; k < K_NEIGH; ++k) {
            v4f v = __builtin_nontemporal_load(base + (size_t)k * (C_DIM / 4));
            s += v;
        }
        s *= (1.0f / (float)K_NEIGH);
        ((v4f*)(g_lds + m * LDS_STRIDE))[c4] = s;
    }
    __syncthreads();

    // ---------------- Phase 2: WMMA GEMM over K = C = 512 ------------------
    // 32 output column tiles of 16; wave w handles tiles {w, w+8, w+16, w+24}.
    const int lane = tid & 31;
    const int wave = tid >> 5;
    const int mrow = lane & 15;          // A: M row / B: N column for this lane
    const int koff = (lane >> 4) * 2;    // lanes 0-15 -> K 0,1 ; lanes 16-31 -> K 2,3
    const int hi   = lane >> 4;

    const float* xrow = x_lds + mrow * LDS_STRIDE + koff;
    const float* grow = g_lds + mrow * LDS_STRIDE + koff;

    #pragma unroll
    for (int i = 0; i < 4; ++i) {
        const int o0   = (wave + 8 * i) * 16;
        const int ocol = o0 + mrow;                       // B column owned by lane
        const float* wl = Wl + (size_t)ocol * C_DIM + koff;
        const float* wr = Wr + (size_t)ocol * C_DIM + koff;

        // Independent accumulators for the two GEMM terms: consecutive WMMAs
        // alternate D targets, so each depends only on the WMMA 2 steps back.
        v8f acc_l = {0.f, 0.f, 0.f, 0.f, 0.f, 0.f, 0.f, 0.f};
        v8f acc_r = {0.f, 0.f, 0.f, 0.f, 0.f, 0.f, 0.f, 0.f};

        // 128 K-steps of 4.
        for (int c0 = 0; c0 < C_DIM; c0 += 4) {
            v2f ax  = *(const v2f*)(xrow + c0);   // A frag of x tile
            v2f ag  = *(const v2f*)(grow + c0);   // A frag of agg tile
            v2f bwl = *(const v2f*)(wl + c0);     // B frag of W_l (K-major pair)
            v2f bwr = *(const v2f*)(wr + c0);     // B frag of W_r
            acc_l = __builtin_amdgcn_wmma_f32_16x16x4_f32(
                        false, ax, false, bwl, (short)0, acc_l, false, false);
            acc_r = __builtin_amdgcn_wmma_f32_16x16x4_f32(
                        false, ag, false, bwr, (short)0, acc_r, false, false);
        }

        // Epilogue: C/D layout -> element (M = v + 8*hi, N = mrow).
        const float bias = bl[ocol] + br[ocol];
        float* orow = out + (size_t)(n0 + hi * 8) * OUT_DIM + ocol;
        #pragma unroll
        for (int v = 0; v < 8; ++v) {
            __builtin_nontemporal_store(acc_l[v] + acc_r[v] + bias,
                                        orow + (size_t)v * OUT_DIM);
        }
    }
}

extern "C" void kernel_launch(void* const* d_in, const int* in_sizes, int n_in,
                              void* d_out, int out_size, void* d_ws, size_t ws_size,
                              hipStream_t stream)
{
    (void)n_in; (void)out_size; (void)d_ws; (void)ws_size;
    const float* x     = (const float*)d_in[0];  // [N, C]
    const float* neigh = (const float*)d_in[1];  // [N, K, C]
    const float* Wl    = (const float*)d_in[2];  // [OUT, C]
    const float* bl    = (const float*)d_in[3];  // [OUT]
    const float* Wr    = (const float*)d_in[4];  // [OUT, C]
    const float* br    = (const float*)d_in[5];  // [OUT]
    float* out = (float*)d_out;                  // [N, OUT]

    const int N = in_sizes[0] / C_DIM;           // 20000, divisible by 16
    const int grid = N / ROWS;                   // 1250 blocks

    sage_fused_kernel<<<grid, 256, 0, stream>>>(x, neigh, Wl, bl, Wr, br, out);
}